// RandLANet_8289286881645
// MI455X (gfx1250) — compile-verified
//
#include <hip/hip_runtime.h>

// ---------------------------------------------------------------------------
// RandLANet forward for gfx1250 (MI455X). wave32, WMMA f32 16x16x4.
//
// Param flattening assumption: d_in[0]=points (1,8192,6) f32,
// d_in[1]=permutation (8192,) i32, then params pytree flattened with dict keys
// sorted alphabetically (JAX convention), lists in order:
//   bn_start{bt,g}, decoder[0..3]{b,bt,g,w}, encoder[0..3]{lse1{b,bt,g,w},
//   lse2{b,bt,g,w}, mlp1{b,w}, mlp2{b,w}, pool1{mlp{b,bt,g,w},score_w},
//   pool2{mlp{b,bt,g,w},score_w}, shortcut{b,bt,g,w}}, fc_end[0..2],
//   fc_start{b,w}, mid{b,w}.
// ---------------------------------------------------------------------------

typedef __attribute__((ext_vector_type(2))) float v2f;
typedef __attribute__((ext_vector_type(8))) float v8f;

#define NPTS 8192
#define KNN_K 16

__device__ __forceinline__ float act_f(float v, int act) {
  if (act == 1) return v > 0.f ? v : 0.f;
  if (act == 2) return v > 0.f ? v : 0.2f * v;
  if (act == 3) return v > 0.f ? v : 0.01f * v;
  return v;
}

// ---------------- coords extraction (permuted, SoA) + squared norms ---------
__global__ void k_coords(const float* __restrict__ pts, const int* __restrict__ perm,
                         float* cx, float* cy, float* cz, float* nr, int n) {
  int i = blockIdx.x * blockDim.x + threadIdx.x;
  if (i >= n) return;
  int s = perm[i];
  float x = pts[s * 6 + 0], y = pts[s * 6 + 1], z = pts[s * 6 + 2];
  cx[i] = x; cy[i] = y; cz[i] = z; nr[i] = x * x + y * y + z * z;
}

// ---------------- fc_start: (N,6) @ W^T(8x6) + b -> channel-major (8,N) -----
__global__ void k_fc_start(const float* __restrict__ pts, const float* __restrict__ W,
                           const float* __restrict__ b, float* __restrict__ X0, int n) {
  int i = blockIdx.x * blockDim.x + threadIdx.x;
  if (i >= n) return;
  float p[6];
#pragma unroll
  for (int c = 0; c < 6; ++c) p[c] = pts[i * 6 + c];
#pragma unroll
  for (int o = 0; o < 8; ++o) {
    float a = b[o];
#pragma unroll
    for (int c = 0; c < 6; ++c) a += W[o * 6 + c] * p[c];
    X0[(size_t)o * n + i] = a;
  }
}

// ---------------- BatchNorm statistics: one block per channel ---------------
__global__ void k_bn_stats(const float* __restrict__ X, int cols, int ld,
                           float* __restrict__ mean, float* __restrict__ rstd) {
  __shared__ float ss[256], sq[256];
  int c = blockIdx.x;
  const float* row = X + (size_t)c * ld;
  float s = 0.f, q = 0.f;
  for (int i = threadIdx.x; i < cols; i += blockDim.x) {
    float v = row[i];
    s += v; q += v * v;
  }
  ss[threadIdx.x] = s; sq[threadIdx.x] = q;
  __syncthreads();
  for (int off = 128; off > 0; off >>= 1) {
    if ((int)threadIdx.x < off) {
      ss[threadIdx.x] += ss[threadIdx.x + off];
      sq[threadIdx.x] += sq[threadIdx.x + off];
    }
    __syncthreads();
  }
  if (threadIdx.x == 0) {
    float m = ss[0] / (float)cols;
    float v = sq[0] / (float)cols - m * m;
    mean[c] = m;
    rstd[c] = rsqrtf(v + 1e-6f);
  }
}

// BN apply (+ optional residual addend, optional gather permutation, act).
__global__ void k_bn_apply(float* __restrict__ Y, const float* __restrict__ X,
                           const float* __restrict__ ADD, const int* __restrict__ gath,
                           const float* __restrict__ mean, const float* __restrict__ rstd,
                           const float* __restrict__ g, const float* __restrict__ bt,
                           int C, int cols, int ldx, int act) {
  long i = blockIdx.x * (long)blockDim.x + threadIdx.x;
  if (i >= (long)C * cols) return;
  int c = (int)(i / cols), col = (int)(i % cols);
  int sc = gath ? gath[col] : col;
  float v = X[(size_t)c * ldx + sc];
  v = (v - mean[c]) * rstd[c] * g[c] + bt[c];
  if (ADD) v += ADD[(size_t)c * cols + col];
  Y[(size_t)c * cols + col] = act_f(v, act);
}

// ---------------- KNN top-16 via V_WMMA_F32_16X16X4_F32 ---------------------
// One wave per 16 query rows. Each 16-ref tile: one WMMA gives the 16x16 dot
// tile (K=4: x,y,z,0). d2 = |q|^2 + |r|^2 - 2*dot. Lanes 0-15 each own one
// query row and keep a register-resident sorted top-16 list.
__global__ __launch_bounds__(32) void k_knn16(
    const float* __restrict__ cx, const float* __restrict__ cy,
    const float* __restrict__ cz, const float* __restrict__ nrm, int n,
    int* __restrict__ oidx, float* __restrict__ odst) {
  __shared__ float tile[16][17];
  __shared__ float rn[16];
  const int lane = threadIdx.x;
  const int lm = lane & 15;
  const bool hi = lane >= 16;
  const int q0 = blockIdx.x * 16;

  // A matrix (16x4 f32): lanes 0-15 -> (K0,K1)=(x,y); lanes 16-31 -> (K2,K3)=(z,0)
  v2f a;
  a.x = hi ? cz[q0 + lm] : cx[q0 + lm];
  a.y = hi ? 0.0f : cy[q0 + lm];

  float best_d[KNN_K];
  int best_i[KNN_K];
#pragma unroll
  for (int t = 0; t < KNN_K; ++t) { best_d[t] = 3.0e38f; best_i[t] = 0; }
  const float qn = nrm[q0 + lm];

  for (int j0 = 0; j0 < n; j0 += 16) {
    // B matrix (4x16 f32): VGPR0 lanes0-15=K0(x), lanes16-31=K2(z); VGPR1 K1(y)/K3(0)
    v2f b;
    b.x = hi ? cz[j0 + lm] : cx[j0 + lm];
    b.y = hi ? 0.0f : cy[j0 + lm];
    v8f c = {};
    c = __builtin_amdgcn_wmma_f32_16x16x4_f32(false, a, false, b, (short)0, c,
                                              false, false);
    // D layout: VGPR v -> row v (lanes 0-15) / row v+8 (lanes 16-31), col = lane%16
    int rbase = hi ? 8 : 0;
#pragma unroll
    for (int v = 0; v < 8; ++v) tile[rbase + v][lm] = c[v];
    if (!hi) rn[lm] = nrm[j0 + lm];
    __syncthreads();
    if (!hi) {
#pragma unroll 1
      for (int col = 0; col < 16; ++col) {
        float d2 = qn + rn[col] - 2.0f * tile[lm][col];
        if (d2 < best_d[KNN_K - 1]) {
          best_d[KNN_K - 1] = d2;
          best_i[KNN_K - 1] = j0 + col;
#pragma unroll
          for (int t = KNN_K - 1; t > 0; --t) {
            if (best_d[t] < best_d[t - 1]) {
              float td = best_d[t]; best_d[t] = best_d[t - 1]; best_d[t - 1] = td;
              int ti = best_i[t]; best_i[t] = best_i[t - 1]; best_i[t - 1] = ti;
            }
          }
        }
      }
    }
    __syncthreads();
  }
  if (!hi) {
#pragma unroll
    for (int t = 0; t < KNN_K; ++t) {
      odst[(size_t)(q0 + lm) * KNN_K + t] = best_d[t];
      oidx[(size_t)(q0 + lm) * KNN_K + t] = best_i[t];
    }
  }
}

// ---------------- 1-NN (decoder upsampling) ---------------------------------
__global__ void k_nn1(const float* __restrict__ cx, const float* __restrict__ cy,
                      const float* __restrict__ cz, int n_q, int n_r,
                      int* __restrict__ out) {
  int q = blockIdx.x * blockDim.x + threadIdx.x;
  if (q >= n_q) return;
  float qx = cx[q], qy = cy[q], qz = cz[q];
  float bd = 3.0e38f; int bi = 0;
  for (int r = 0; r < n_r; ++r) {
    float dx = qx - cx[r], dy = qy - cy[r], dz = qz - cz[r];
    float d = dx * dx + dy * dy + dz * dz;
    if (d < bd) { bd = d; bi = r; }
  }
  out[q] = bi;
}

// ---------------- geometric encoding (10 x n*16) ----------------------------
__global__ void k_geo(const float* __restrict__ cx, const float* __restrict__ cy,
                      const float* __restrict__ cz, const int* __restrict__ idx,
                      const float* __restrict__ dst, float* __restrict__ G, int n) {
  int i = blockIdx.x * blockDim.x + threadIdx.x;
  size_t cols = (size_t)n * 16;
  if (i >= (int)cols) return;
  int pt = i >> 4;
  int nb = idx[i];
  float px = cx[pt], py = cy[pt], pz = cz[pt];
  float nx = cx[nb], ny = cy[nb], nz = cz[nb];
  G[0 * cols + i] = px; G[1 * cols + i] = py; G[2 * cols + i] = pz;
  G[3 * cols + i] = nx; G[4 * cols + i] = ny; G[5 * cols + i] = nz;
  G[6 * cols + i] = px - nx; G[7 * cols + i] = py - ny; G[8 * cols + i] = pz - nz;
  G[9 * cols + i] = dst[i];
}

// broadcast feats (h x n) into xcat rows [h,2h) over k=16
__global__ void k_broadcast(const float* __restrict__ X, float* __restrict__ XC,
                            int h, int n) {
  long i = blockIdx.x * (long)blockDim.x + threadIdx.x;
  long cols = (long)n * 16;
  if (i >= (long)h * cols) return;
  int c = (int)(i / cols);
  long col = i % cols;
  XC[(size_t)(h + c) * cols + col] = X[(size_t)c * n + (col >> 4)];
}

// attention pool: softmax over k of Z, weighted sum of XC -> F (C x n)
__global__ void k_attpool(const float* __restrict__ Z, const float* __restrict__ XC,
                          float* __restrict__ F, int C, int n) {
  long i = blockIdx.x * (long)blockDim.x + threadIdx.x;
  if (i >= (long)C * n) return;
  int o = (int)(i / n), pt = (int)(i % n);
  const float* z = Z + (size_t)o * n * 16 + (size_t)pt * 16;
  const float* x = XC + (size_t)o * n * 16 + (size_t)pt * 16;
  float m = -3.0e38f;
#pragma unroll
  for (int k = 0; k < 16; ++k) m = fmaxf(m, z[k]);
  float e[16];
  float s = 0.f;
#pragma unroll
  for (int k = 0; k < 16; ++k) { e[k] = __expf(z[k] - m); s += e[k]; }
  float inv = 1.0f / s;
  float acc = 0.f;
#pragma unroll
  for (int k = 0; k < 16; ++k) acc += e[k] * inv * x[k];
  F[(size_t)o * n + pt] = acc;
}

// ---------------- GEMM: Y(cout x cols) = W(cout x cin) @ X(cin x cols) ------
__global__ void k_gemm_scalar(const float* __restrict__ W, const float* __restrict__ X,
                              const float* __restrict__ bias, float* __restrict__ Y,
                              int cout, int cin, int cols, int ldx, int act) {
  long i = blockIdx.x * (long)blockDim.x + threadIdx.x;
  if (i >= (long)cout * cols) return;
  int o = (int)(i / cols), col = (int)(i % cols);
  float acc = bias ? bias[o] : 0.f;
  const float* w = W + (size_t)o * cin;
  for (int c = 0; c < cin; ++c) acc += w[c] * X[(size_t)c * ldx + col];
  Y[(size_t)o * cols + col] = act_f(acc, act);
}

// WMMA f32 GEMM: one wave per 16x16 output tile, K stepped by 4.
__global__ __launch_bounds__(32) void k_gemm_wmma(
    const float* __restrict__ W, const float* __restrict__ X,
    const float* __restrict__ bias, float* __restrict__ Y, int cout, int cin,
    int cols, int ldx, int act) {
  const int lane = threadIdx.x;
  const int lm = lane & 15;
  const bool hi = lane >= 16;
  const int m0 = blockIdx.y * 16;
  const int col0 = blockIdx.x * 16;
  const int col = col0 + lm;
  const int colc = col < cols ? col : (cols - 1);
  const int k01 = hi ? 2 : 0;
  const int k23 = hi ? 3 : 1;
  const float* wrow = W + (size_t)(m0 + lm) * cin;

  v8f acc = {};
  for (int kk = 0; kk < cin; kk += 4) {
    v2f a, b;
    a.x = wrow[kk + k01];
    a.y = wrow[kk + k23];
    b.x = X[(size_t)(kk + k01) * ldx + colc];
    b.y = X[(size_t)(kk + k23) * ldx + colc];
    acc = __builtin_amdgcn_wmma_f32_16x16x4_f32(false, a, false, b, (short)0,
                                                acc, false, false);
  }
  if (col < cols) {
#pragma unroll
    for (int v = 0; v < 8; ++v) {
      int row = m0 + v + (hi ? 8 : 0);
      float y = acc[v] + (bias ? bias[row] : 0.f);
      Y[(size_t)row * cols + col] = act_f(y, act);
    }
  }
}

// ---------------- gather / copy / scatter helpers ---------------------------
__global__ void k_gather_cols(const float* __restrict__ X, const int* __restrict__ idx,
                              float* __restrict__ Y, int C, int n_out, int ldx, int ldy) {
  long i = blockIdx.x * (long)blockDim.x + threadIdx.x;
  if (i >= (long)C * n_out) return;
  int c = (int)(i / n_out), q = (int)(i % n_out);
  Y[(size_t)c * ldy + q] = X[(size_t)c * ldx + idx[q]];
}

__global__ void k_copy_mat(const float* __restrict__ X, float* __restrict__ Y,
                           int C, int cols, int ldx, int ldy) {
  long i = blockIdx.x * (long)blockDim.x + threadIdx.x;
  if (i >= (long)C * cols) return;
  int c = (int)(i / cols), col = (int)(i % cols);
  Y[(size_t)c * ldy + col] = X[(size_t)c * ldx + col];
}

__global__ void k_scatter_cols(const float* __restrict__ X, const int* __restrict__ perm,
                               float* __restrict__ Y, int C, int n) {
  long i = blockIdx.x * (long)blockDim.x + threadIdx.x;
  if (i >= (long)C * n) return;
  int c = (int)(i / n), col = (int)(i % n);
  Y[(size_t)c * n + perm[col]] = X[(size_t)c * n + col];
}

// ===========================================================================
extern "C" void kernel_launch(void* const* d_in, const int* in_sizes, int n_in,
                              void* d_out, int out_size, void* d_ws, size_t ws_size,
                              hipStream_t stream) {
  (void)in_sizes; (void)n_in; (void)out_size; (void)ws_size;
  const float* pts = (const float*)d_in[0];
  const int* perm = (const int*)d_in[1];
  auto P = [&](int i) -> const float* { return (const float*)d_in[i]; };

  // ---- param index map (see header comment) ----
  // bn_start: bt=2, g=3 ; decoder i: base 4+4i (b,bt,g,w)
  // encoder l: base 20+26l; offsets:
  //   0..3 lse1{b,bt,g,w}, 4..7 lse2, 8..9 mlp1{b,w}, 10..11 mlp2{b,w},
  //   12..16 pool1{mlp{b,bt,g,w},score_w}, 17..21 pool2, 22..25 shortcut
  // fc_end: 124..127, 128..131, 132..133(b,w); fc_start b=134,w=135; mid b=136,w=137

  // ---- workspace partition (floats) ----
  float* w = (float*)d_ws;
  size_t off = 0;
  auto alloc = [&](size_t nf) -> float* { float* p = w + off; off += nf; return p; };
  float* CX = alloc(NPTS);  float* CY = alloc(NPTS);
  float* CZ = alloc(NPTS);  float* NRM = alloc(NPTS);
  float* MEAN = alloc(512); float* RSTD = alloc(512);
  int* KIDX = (int*)alloc(NPTS * KNN_K);
  float* KDST = alloc(NPTS * KNN_K);
  int* NN1 = (int*)alloc(NPTS);
  float* GEO = alloc((size_t)10 * NPTS * KNN_K);       // 1.31M
  float* XCAT = alloc((size_t)16 * NPTS * KNN_K);      // 2.10M
  float* Z = alloc((size_t)16 * NPTS * KNN_K);         // 2.10M
  float* BUF0 = alloc(1048576);
  float* BUF1 = alloc(1048576);
  float* BUF2 = alloc(1048576);
  float* BUF3 = alloc(1048576);
  float* stacks[4] = { alloc(32 * 8192), alloc(128 * 2048),
                       alloc(256 * 512), alloc(512 * 128) };

  auto gemm = [&](const float* W_, const float* X_, const float* b_, float* Y_,
                  int cout, int cin, int cols, int ldx, int act) {
    if ((cout % 16 == 0) && (cin % 4 == 0) && cols >= 16) {
      dim3 g((cols + 15) / 16, cout / 16);
      k_gemm_wmma<<<g, 32, 0, stream>>>(W_, X_, b_, Y_, cout, cin, cols, ldx, act);
    } else {
      long total = (long)cout * cols;
      k_gemm_scalar<<<dim3((unsigned)((total + 255) / 256)), 256, 0, stream>>>(
          W_, X_, b_, Y_, cout, cin, cols, ldx, act);
    }
  };
  auto bn = [&](float* Y, const float* X, const float* ADD, const int* gath,
                const float* g_, const float* bt_, int C, int cols, int ldx, int act) {
    k_bn_stats<<<C, 256, 0, stream>>>(X, cols, ldx, MEAN, RSTD);
    long total = (long)C * cols;
    k_bn_apply<<<dim3((unsigned)((total + 255) / 256)), 256, 0, stream>>>(
        Y, X, ADD, gath, MEAN, RSTD, g_, bt_, C, cols, ldx, act);
  };
  auto blocks = [](long total) { return dim3((unsigned)((total + 255) / 256)); };

  // ---- fc_start + bn_start + leaky(0.2) + permutation ----
  k_coords<<<blocks(NPTS), 256, 0, stream>>>(pts, perm, CX, CY, CZ, NRM, NPTS);
  k_fc_start<<<blocks(NPTS), 256, 0, stream>>>(pts, P(135), P(134), BUF1, NPTS);
  bn(BUF0, BUF1, nullptr, perm, P(3), P(2), 8, NPTS, NPTS, 2);  // -> x (8 x 8192)

  // ---- encoder ----
  const int dinv[4] = { 8, 32, 128, 256 };
  const int doutv[4] = { 16, 64, 128, 256 };
  int n = NPTS;
  for (int l = 0; l < 4; ++l) {
    const int din = dinv[l], dout = doutv[l], h = dout / 2;
    const int cols16 = n * 16;
    const int eb = 20 + 26 * l;
    k_knn16<<<n / 16, 32, 0, stream>>>(CX, CY, CZ, NRM, n, KIDX, KDST);
    // mlp1 (leaky 0.2)
    gemm(P(eb + 9), BUF0, P(eb + 8), BUF1, h, din, n, n, 2);
    k_geo<<<blocks(cols16), 256, 0, stream>>>(CX, CY, CZ, KIDX, KDST, GEO, n);
    // lse1 -> xcat[0,h), BN+relu
    gemm(P(eb + 3), GEO, P(eb + 0), XCAT, h, 10, cols16, cols16, 0);
    bn(XCAT, XCAT, nullptr, nullptr, P(eb + 2), P(eb + 1), h, cols16, cols16, 1);
    k_broadcast<<<blocks((long)h * cols16), 256, 0, stream>>>(BUF1, XCAT, h, n);
    // pool1: scores (no bias), softmax-weighted sum, mlp + BN + relu
    gemm(P(eb + 16), XCAT, nullptr, Z, dout, dout, cols16, cols16, 0);
    k_attpool<<<blocks((long)dout * n), 256, 0, stream>>>(Z, XCAT, BUF2, dout, n);
    gemm(P(eb + 15), BUF2, P(eb + 12), BUF3, h, dout, n, n, 0);
    bn(BUF3, BUF3, nullptr, nullptr, P(eb + 14), P(eb + 13), h, n, n, 1);
    // lse2 + pool2
    gemm(P(eb + 7), GEO, P(eb + 4), XCAT, h, 10, cols16, cols16, 0);
    bn(XCAT, XCAT, nullptr, nullptr, P(eb + 6), P(eb + 5), h, cols16, cols16, 1);
    k_broadcast<<<blocks((long)h * cols16), 256, 0, stream>>>(BUF3, XCAT, h, n);
    gemm(P(eb + 21), XCAT, nullptr, Z, dout, dout, cols16, cols16, 0);
    k_attpool<<<blocks((long)dout * n), 256, 0, stream>>>(Z, XCAT, BUF2, dout, n);
    gemm(P(eb + 20), BUF2, P(eb + 17), BUF1, dout, dout, n, n, 0);
    bn(BUF1, BUF1, nullptr, nullptr, P(eb + 19), P(eb + 18), dout, n, n, 1);
    // mlp2 + BN(shortcut) -> leaky(0.01) -> stack
    gemm(P(eb + 11), BUF1, P(eb + 10), BUF2, 2 * dout, dout, n, n, 0);
    gemm(P(eb + 25), BUF0, P(eb + 22), BUF3, 2 * dout, din, n, n, 0);
    bn(stacks[l], BUF3, BUF2, nullptr, P(eb + 24), P(eb + 23), 2 * dout, n, n, 3);
    // downsample prefix for next level
    const int nn2 = n / 4;
    k_copy_mat<<<blocks((long)2 * dout * nn2), 256, 0, stream>>>(
        stacks[l], BUF0, 2 * dout, nn2, n, nn2);
    n = nn2;
  }

  // ---- mid: 512x512 @ (512 x 32), relu ----
  gemm(P(137), BUF0, P(136), BUF1, 512, 512, 32, 32, 1);

  // ---- decoder ----
  const int dec_cout[4] = { 256, 128, 32, 8 };
  float* xc = BUF1;
  float* xo = BUF3;
  int Ccur = 512;
  int n_r = 32;
  for (int i = 0; i < 4; ++i) {
    const int n_q = 4 * n_r;
    const int Cs = 2 * doutv[3 - i];
    const int db = 4 + 4 * i;
    k_nn1<<<blocks(n_q), 256, 0, stream>>>(CX, CY, CZ, n_q, n_r, NN1);
    k_gather_cols<<<blocks((long)Ccur * n_q), 256, 0, stream>>>(
        xc, NN1, BUF2, Ccur, n_q, n_r, n_q);
    k_copy_mat<<<blocks((long)Cs * n_q), 256, 0, stream>>>(
        stacks[3 - i], BUF2 + (size_t)Ccur * n_q, Cs, n_q, n_q, n_q);
    gemm(P(db + 3), BUF2, P(db + 0), xo, dec_cout[i], Ccur + Cs, n_q, n_q, 0);
    bn(xo, xo, nullptr, nullptr, P(db + 2), P(db + 1), dec_cout[i], n_q, n_q, 1);
    float* t = xc; xc = xo; xo = t;
    Ccur = dec_cout[i];
    n_r = n_q;
  }

  // ---- inverse permutation: out[c][perm[i]] = x[c][i] ----
  k_scatter_cols<<<blocks((long)8 * NPTS), 256, 0, stream>>>(xc, perm, BUF0, 8, NPTS);

  // ---- fc_end ----
  gemm(P(127), BUF0, P(124), BUF2, 64, 8, NPTS, NPTS, 0);
  bn(BUF2, BUF2, nullptr, nullptr, P(126), P(125), 64, NPTS, NPTS, 1);
  gemm(P(131), BUF2, P(128), BUF0, 32, 64, NPTS, NPTS, 0);
  bn(BUF0, BUF0, nullptr, nullptr, P(130), P(129), 32, NPTS, NPTS, 1);
  gemm(P(133), BUF0, P(132), (float*)d_out, 13, 32, NPTS, NPTS, 0);
}